// Mixhop_91207925498526
// MI455X (gfx1250) — compile-verified
//
#include <hip/hip_runtime.h>
#include <hip/hip_bf16.h>
#include <math.h>

typedef __attribute__((ext_vector_type(2))) float v2f;
typedef __attribute__((ext_vector_type(8))) float v8f;

#define FEAT_IN 512
#define HID     128
#define NCLS    40

__device__ __forceinline__ v8f wmma_f32_k4(v2f a, v2f b, v8f c) {
  // D = A(16x4 f32) * B(4x16 f32) + C(16x16 f32)
  return __builtin_amdgcn_wmma_f32_16x16x4_f32(
      /*neg_a=*/false, a, /*neg_b=*/false, b,
      /*c_mod=*/(short)0, c, /*reuse_a=*/false, /*reuse_b=*/false);
}

// ---------------- degree / dinv ----------------
__global__ void deg_init(float* deg, int n) {
  int i = blockIdx.x * blockDim.x + threadIdx.x;
  if (i < n) deg[i] = 1.0f;   // self loop
}

__global__ void deg_accum(const int* __restrict__ dst, float* deg, int e) {
  int i = blockIdx.x * blockDim.x + threadIdx.x;
  if (i < e) atomicAdd(&deg[dst[i]], 1.0f);
}

__global__ void deg_rsqrt(float* d, int n) {
  int i = blockIdx.x * blockDim.x + threadIdx.x;
  if (i < n) d[i] = rsqrtf(d[i]);   // deg >= 1 always
}

// ---------------- GEMM1: h0 = x @ W1 + b1 (fp32 WMMA) ----------------
// One wave per 16-row M-tile; computes all 128 output columns (8 N-tiles).
__global__ __launch_bounds__(256)
void gemm1_wmma(const float* __restrict__ x, const float* __restrict__ W1,
                const float* __restrict__ b1, float* __restrict__ h0) {
  const int lane  = threadIdx.x & 31;
  const int wave  = threadIdx.x >> 5;
  const int mtile = blockIdx.x * 8 + wave;
  const int l15   = lane & 15;
  const int khalf = lane >> 4;           // 0: K=0,1  1: K=2,3
  const int row   = mtile * 16 + l15;
  const float* xrow = x + (size_t)row * FEAT_IN;

  v8f acc[8];
  #pragma unroll
  for (int t = 0; t < 8; ++t) {
    float bv = b1[t * 16 + l15];         // bias depends only on N-column
    #pragma unroll
    for (int r = 0; r < 8; ++r) acc[t][r] = bv;
  }

  for (int k = 0; k < FEAT_IN; k += 4) {
    const int ka = k + khalf * 2;
    v2f a = *(const v2f*)(xrow + ka);    // A[row][ka], A[row][ka+1]
    const float* w0 = W1 + (size_t)ka * HID;
    #pragma unroll
    for (int t = 0; t < 8; ++t) {
      const int col = t * 16 + l15;
      v2f b;
      b.x = w0[col];                     // B[ka][col]
      b.y = w0[HID + col];               // B[ka+1][col]
      acc[t] = wmma_f32_k4(a, b, acc[t]);
    }
  }

  #pragma unroll
  for (int t = 0; t < 8; ++t) {
    const int col = t * 16 + l15;
    #pragma unroll
    for (int r = 0; r < 8; ++r) {
      const int m = r + khalf * 8;
      h0[(size_t)(mtile * 16 + m) * HID + col] = acc[t][r];
    }
  }
}

// ---------------- propagation ----------------
// self loop term: h_out[i] = dinv[i]^2 * h_in[i]
__global__ void self_loop_init(const float* __restrict__ dinv,
                               const float* __restrict__ hin,
                               float* __restrict__ hout, int total) {
  int idx = blockIdx.x * blockDim.x + threadIdx.x;
  if (idx >= total) return;
  int node = idx >> 7;                   // /128
  float w = dinv[node];
  hout[idx] = w * w * hin[idx];
}

// one wave per edge; 4 features per lane (128/32); fp32 scatter-atomics to L2
__global__ __launch_bounds__(256)
void edge_prop(const int* __restrict__ src, const int* __restrict__ dst,
               const float* __restrict__ dinv, const float* __restrict__ hin,
               float* __restrict__ hout, int e) {
  int tid  = blockIdx.x * blockDim.x + threadIdx.x;
  int edge = tid >> 5;
  if (edge >= e) return;
  int lane = tid & 31;
  int s = src[edge];
  int d = dst[edge];
  float w = dinv[s] * dinv[d];
  const float4 hv = *(const float4*)(hin + (size_t)s * HID + lane * 4);
  float* o = hout + (size_t)d * HID + lane * 4;
  atomicAdd(o + 0, w * hv.x);
  atomicAdd(o + 1, w * hv.y);
  atomicAdd(o + 2, w * hv.z);
  atomicAdd(o + 3, w * hv.w);
}

// ---------------- GEMM2 + bias + log_softmax ----------------
// logits = concat(relu(h0),relu(h1),relu(h2)) @ W2 + b2 ; out = log_softmax
// One wave per 16-row tile; 3 column tiles cover 40 classes (cols 40..47 masked).
__global__ __launch_bounds__(256)
void gemm2_logsoftmax(const float* __restrict__ h0, const float* __restrict__ h1,
                      const float* __restrict__ h2, const float* __restrict__ W2,
                      const float* __restrict__ b2, float* __restrict__ out) {
  const int lane  = threadIdx.x & 31;
  const int wave  = threadIdx.x >> 5;
  const int mtile = blockIdx.x * 8 + wave;
  const int l15   = lane & 15;
  const int khalf = lane >> 4;
  const int row   = mtile * 16 + l15;

  v8f acc[3];
  #pragma unroll
  for (int t = 0; t < 3; ++t)
    #pragma unroll
    for (int r = 0; r < 8; ++r) acc[t][r] = 0.0f;

  #pragma unroll 1
  for (int seg = 0; seg < 3; ++seg) {
    const float* hseg = (seg == 0) ? h0 : ((seg == 1) ? h1 : h2);
    const float* hrow = hseg + (size_t)row * HID;
    for (int kk = 0; kk < HID; kk += 4) {
      const int ka = kk + khalf * 2;
      v2f a;
      a.x = fmaxf(hrow[ka], 0.0f);       // relu fused into A read
      a.y = fmaxf(hrow[ka + 1], 0.0f);
      const int kb = seg * HID + ka;     // row of W2 (0..383)
      const float* w0 = W2 + (size_t)kb * NCLS;
      #pragma unroll
      for (int t = 0; t < 3; ++t) {
        const int col  = t * 16 + l15;
        const int colc = (col < NCLS) ? col : (NCLS - 1);  // clamp: no OOB load
        const bool ok  = (col < NCLS);
        v2f b;
        b.x = ok ? w0[colc] : 0.0f;
        b.y = ok ? w0[NCLS + colc] : 0.0f;
        acc[t] = wmma_f32_k4(a, b, acc[t]);
      }
    }
  }

  // epilogue: per output row (VGPR r, lane half), 40-way log-softmax.
  // xor masks 1,2,4,8 stay inside each 16-lane group => never mix rows.
  #pragma unroll
  for (int r = 0; r < 8; ++r) {
    float v[3];
    #pragma unroll
    for (int t = 0; t < 3; ++t) {
      const int col = t * 16 + l15;
      v[t] = (col < NCLS) ? (acc[t][r] + b2[col]) : -__builtin_inff();
    }
    float mx = fmaxf(fmaxf(v[0], v[1]), v[2]);
    #pragma unroll
    for (int off = 1; off <= 8; off <<= 1)
      mx = fmaxf(mx, __shfl_xor(mx, off, 32));
    float s = 0.0f;
    #pragma unroll
    for (int t = 0; t < 3; ++t) {
      const int col = t * 16 + l15;
      if (col < NCLS) s += expf(v[t] - mx);
    }
    #pragma unroll
    for (int off = 1; off <= 8; off <<= 1)
      s += __shfl_xor(s, off, 32);
    const float lse = mx + logf(s);
    const int m = r + khalf * 8;
    float* orow = out + (size_t)(mtile * 16 + m) * NCLS;
    #pragma unroll
    for (int t = 0; t < 3; ++t) {
      const int col = t * 16 + l15;
      if (col < NCLS) orow[col] = v[t] - lse;
    }
  }
}

extern "C" void kernel_launch(void* const* d_in, const int* in_sizes, int n_in,
                              void* d_out, int out_size, void* d_ws, size_t ws_size,
                              hipStream_t stream) {
  const float* x  = (const float*)d_in[0];
  const int*   ei = (const int*)d_in[1];
  const float* W1 = (const float*)d_in[2];
  const float* b1 = (const float*)d_in[3];
  const float* W2 = (const float*)d_in[4];
  const float* b2 = (const float*)d_in[5];
  float* out = (float*)d_out;

  const int N = in_sizes[0] / FEAT_IN;   // 65536 nodes
  const int E = in_sizes[1] / 2;         // 1048576 edges
  const int* src = ei;                   // edge_index[0]
  const int* dst = ei + E;               // edge_index[1]

  // workspace: dinv | h0 | h1 | h2
  char* ws = (char*)d_ws;
  float* dinv = (float*)ws;
  size_t off = ((size_t)N * sizeof(float) + 255) & ~(size_t)255;
  float* h0 = (float*)(ws + off);
  float* h1 = h0 + (size_t)N * HID;
  float* h2 = h1 + (size_t)N * HID;

  // 1) GCN normalization
  deg_init  <<<(N + 255) / 256, 256, 0, stream>>>(dinv, N);
  deg_accum <<<(E + 255) / 256, 256, 0, stream>>>(dst, dinv, E);
  deg_rsqrt <<<(N + 255) / 256, 256, 0, stream>>>(dinv, N);

  // 2) h0 = x @ W1 + b1  (fp32 WMMA; 4096 M-tiles, 8 waves/block)
  gemm1_wmma<<<N / 128, 256, 0, stream>>>(x, W1, b1, h0);

  // 3) two propagation hops (pre-ReLU h is propagated, per reference)
  const int TOT = N * HID;
  self_loop_init<<<TOT / 256, 256, 0, stream>>>(dinv, h0, h1, TOT);
  edge_prop<<<(E * 32) / 256, 256, 0, stream>>>(src, dst, dinv, h0, h1, E);
  self_loop_init<<<TOT / 256, 256, 0, stream>>>(dinv, h1, h2, TOT);
  edge_prop<<<(E * 32) / 256, 256, 0, stream>>>(src, dst, dinv, h1, h2, E);

  // 4) fused GEMM2 + bias + log_softmax
  gemm2_logsoftmax<<<N / 128, 256, 0, stream>>>(h0, h1, h2, W2, b2, out);
}